// CapacityCalibratedPartialTransportReranker_15556371546810
// MI455X (gfx1250) — compile-verified
//
#include <hip/hip_runtime.h>
#include <math.h>

typedef __attribute__((ext_vector_type(2))) float v2f;
typedef __attribute__((ext_vector_type(8))) float v8f;

#define M_ 1024
#define N_ 4096
#define D_ 128
#define MN (M_ * N_)
#define EPS_ 0.05f
#define ITERS_ 50
#define TOL_ 1e-9f
#define STRENGTH_ 0.5f
#define TINY_ 1e-12f

// scalar slots in ws (index into float/uint array at OFF_SC)
#define S_CMAX 0   // uint bits of max raw cost (>=0, bit-order == float-order)
#define S_SUMK 1   // float sum of Gibbs kernel
#define S_T 2      // float total transported mass this iter
#define S_DELTA 3  // uint bits of max |p - plan| (>=0)
#define S_DONE 4   // uint 0/1 convergence freeze flag
#define S_BMAX 5   // float max(target_capacity)

// ws layout (float offsets)
#define OFF_SC 0
#define OFF_ROW 16                   // rowscale [M]
#define OFF_COL (OFF_ROW + M_)       // colscale [N]
#define OFF_U2 (OFF_COL + N_)        // |u|^2 [M]
#define OFF_V2 (OFF_U2 + M_)         // |v|^2 [N]
#define OFF_PART (OFF_V2 + N_)       // partials [8*N_ = 32768]
#define OFF_C (OFF_PART + 32768)     // cost matrix C [MN]
#define OFF_P0 (OFF_C + MN)          // plan ping buffer [MN]

__global__ void k_init(float* sc) {
  unsigned* u = (unsigned*)sc;
  u[S_CMAX] = 0u;
  u[S_DELTA] = 0x7f800000u;  // +inf: iteration 0 must not trip the freeze
  u[S_DONE] = 0u;
  sc[S_SUMK] = 0.0f;
  sc[S_T] = 0.0f;
  sc[S_BMAX] = 0.0f;
}

// squared row norms: one wave32 per row, rows [0,M) = user, [M, M+N) = item
__global__ void k_sqnorm(const float* __restrict__ user, const float* __restrict__ item,
                         float* __restrict__ u2, float* __restrict__ v2) {
  int wave = (blockIdx.x * blockDim.x + threadIdx.x) >> 5;
  int lane = threadIdx.x & 31;
  const float* src;
  float* dst;
  if (wave < M_) { src = user + (size_t)wave * D_; dst = u2 + wave; }
  else           { src = item + (size_t)(wave - M_) * D_; dst = v2 + (wave - M_); }
  float4 q = ((const float4*)src)[lane];  // 32 lanes * 4 = 128 elems
  float s = q.x * q.x + q.y * q.y + q.z * q.z + q.w * q.w;
#pragma unroll
  for (int off = 16; off >= 1; off >>= 1) s += __shfl_xor(s, off, 32);
  if (lane == 0) *dst = s;
}

// max of target_capacity (4096 values), single block
__global__ void k_bmax(const float* __restrict__ b, float* sc) {
  __shared__ float sm[1024];
  int t = threadIdx.x;
  float m = fmaxf(fmaxf(b[t], b[t + 1024]), fmaxf(b[t + 2048], b[t + 3072]));
  sm[t] = m;
  __syncthreads();
  for (int s = 512; s > 0; s >>= 1) {
    if (t < s) sm[t] = fmaxf(sm[t], sm[t + s]);
    __syncthreads();
  }
  if (t == 0) sc[S_BMAX] = sm[0];
}

// WMMA cost GEMM: raw[m][n] = max(|u_m|^2 + |v_n|^2 - 2 u_m.v_n, 0); track global max.
// One wave per 16x16 tile; K-loop of 32 x V_WMMA_F32_16X16X4_F32.
__global__ void k_gemm_cost(const float* __restrict__ A, const float* __restrict__ Bt,
                            const float* __restrict__ u2, const float* __restrict__ v2,
                            float* __restrict__ Cst, float* sc) {
  int wave = (blockIdx.x * blockDim.x + threadIdx.x) >> 5;
  int lane = threadIdx.x & 31;
  int tm = wave >> 8;    // 64 M-tiles
  int tn = wave & 255;   // 256 N-tiles
  int half = lane >> 4;  // K sub-pair / row-half select
  int l16 = lane & 15;
  const float* arow = A + (size_t)(tm * 16 + l16) * D_;   // A: lane holds row m=l16, K pair
  const float* brow = Bt + (size_t)(tn * 16 + l16) * D_;  // B=item^T: lane holds col n=l16
  v8f acc = {};
#pragma unroll 4
  for (int k = 0; k < D_; k += 4) {
    v2f a = *(const v2f*)(arow + k + 2 * half);  // A[m][k+2h], A[m][k+2h+1]
    v2f b = *(const v2f*)(brow + k + 2 * half);  // B[k+2h][n], B[k+2h+1][n]
    acc = __builtin_amdgcn_wmma_f32_16x16x4_f32(false, a, false, b, (short)0, acc,
                                                false, false);
  }
  float wmax = 0.0f;
#pragma unroll
  for (int v = 0; v < 8; ++v) {
    int m = tm * 16 + v + 8 * half;  // C/D layout: VGPR v -> rows v (lanes 0-15), v+8 (16-31)
    int n = tn * 16 + l16;
    float raw = fmaxf(u2[m] + v2[n] - 2.0f * acc[v], 0.0f);
    Cst[(size_t)m * N_ + n] = raw;
    wmax = fmaxf(wmax, raw);
  }
#pragma unroll
  for (int off = 16; off >= 1; off >>= 1) wmax = fmaxf(wmax, __shfl_xor(wmax, off, 32));
  if (lane == 0) atomicMax((unsigned*)&sc[S_CMAX], __float_as_uint(wmax));
}

// C = raw/(cmax+TINY) + STRENGTH*(1 - b[n]/(bmax+TINY));  K = exp(-C/EPS) -> plan buffer
__global__ void k_kernelize(float* __restrict__ Cst, float* __restrict__ P,
                            const float* __restrict__ b, const float* sc,
                            float* __restrict__ part) {
  __shared__ float sm[256];
  int t = threadIdx.x;
  float cmax = __uint_as_float(((const unsigned*)sc)[S_CMAX]);
  float bmax = sc[S_BMAX];
  float inv_cmax = 1.0f / (cmax + TINY_);
  float inv_bmax = 1.0f / (bmax + TINY_);
  int base = blockIdx.x * 1024;
  float acc = 0.0f;
#pragma unroll
  for (int i = 0; i < 4; ++i) {
    int e = base + i * 256 + t;
    int n = e & (N_ - 1);
    float Cv = Cst[e] * inv_cmax + STRENGTH_ * (1.0f - b[n] * inv_bmax);
    Cst[e] = Cv;
    float Kv = __expf(-Cv / EPS_);
    P[e] = Kv;
    acc += Kv;
  }
  sm[t] = acc;
  __syncthreads();
  for (int s = 128; s > 0; s >>= 1) {
    if (t < s) sm[t] += sm[t + s];
    __syncthreads();
  }
  if (t == 0) part[blockIdx.x] = sm[0];
}

// deterministic fixed-order reduction of 4096 partials; out[0] = mul * sum
__global__ void k_reduce4096(const float* __restrict__ part, float* out, float mul) {
  __shared__ float sm[1024];
  int t = threadIdx.x;
  sm[t] = part[t] + part[t + 1024] + part[t + 2048] + part[t + 3072];
  __syncthreads();
  for (int s = 512; s > 0; s >>= 1) {
    if (t < s) sm[t] += sm[t + s];
    __syncthreads();
  }
  if (t == 0) out[0] = mul * sm[0];
}

// plan0 = K * budget / (sumK + TINY)
__global__ void k_scale0(float* __restrict__ P, const float* sc,
                         const float* __restrict__ budget) {
  float s = budget[0] / (sc[S_SUMK] + TINY_);
  int base = blockIdx.x * 1024 + threadIdx.x;
#pragma unroll
  for (int i = 0; i < 4; ++i) P[base + i * 256] *= s;
}

// row sums -> rowscale[m] = min(a[m]/(r+TINY), 1); one block per row
__global__ void k_rowscale(const float* __restrict__ P, const float* __restrict__ a,
                           float* __restrict__ rowscale) {
  __shared__ float sm[256];
  int m = blockIdx.x, t = threadIdx.x;
  const float* row = P + (size_t)m * N_;
  float s = 0.0f;
#pragma unroll 4
  for (int i = t; i < N_; i += 256) s += row[i];
  sm[t] = s;
  __syncthreads();
  for (int st = 128; st > 0; st >>= 1) {
    if (t < st) sm[t] += sm[t + st];
    __syncthreads();
  }
  if (t == 0) rowscale[m] = fminf(a[m] / (sm[0] + TINY_), 1.0f);
}

// column partial sums over 128-row chunks (8 chunks), coalesced over n
__global__ void k_colpart(const float* __restrict__ P, const float* __restrict__ rowscale,
                          float* __restrict__ part) {
  int chunk = blockIdx.x >> 4;             // 0..7
  int n = (blockIdx.x & 15) * 256 + threadIdx.x;
  int m0 = chunk * 128;
  float s = 0.0f;
#pragma unroll 4
  for (int m = m0; m < m0 + 128; ++m) s += P[(size_t)m * N_ + n] * rowscale[m];
  part[chunk * N_ + n] = s;
}

// combine chunks -> c[n]; colscale[n] = min(b/c,1); part[n] = c*cs (col total after clip)
__global__ void k_colfinish(float* __restrict__ part, const float* __restrict__ b,
                            float* __restrict__ colscale, float* __restrict__ usage_out) {
  int n = blockIdx.x * 256 + threadIdx.x;
  float c = 0.0f;
#pragma unroll
  for (int ch = 0; ch < 8; ++ch) c += part[ch * N_ + n];
  float cs = fminf(b[n] / (c + TINY_), 1.0f);
  colscale[n] = cs;
  float cm = c * cs;
  part[n] = cm;  // thread n only touches part[n] and reads part[ch*N_+n]: no hazard
  if (usage_out) usage_out[n] = cm;
}

// t = sum of clipped column totals; thread 0 also applies last iter's freeze update
__global__ void k_tsum(const float* __restrict__ part, float* sc) {
  __shared__ float sm[1024];
  int t = threadIdx.x;
  sm[t] = part[t] + part[t + 1024] + part[t + 2048] + part[t + 3072];
  __syncthreads();
  for (int s = 512; s > 0; s >>= 1) {
    if (t < s) sm[t] += sm[t + s];
    __syncthreads();
  }
  if (t == 0) {
    sc[S_T] = sm[0];
    unsigned* u = (unsigned*)sc;
    float delta = __uint_as_float(u[S_DELTA]);   // from previous iteration's update
    if (delta < TOL_) u[S_DONE] = 1u;            // done |= (delta < TOL)
    u[S_DELTA] = 0u;                             // reset accumulator for this iteration
  }
}

// p = plan*rs*cs*(budget/(t+TINY)); delta = max|p-plan|; out = done ? plan : p
__global__ void k_update(const float* __restrict__ Pin, float* __restrict__ Pout,
                         const float* __restrict__ rowscale, const float* __restrict__ colscale,
                         float* sc, const float* __restrict__ budget) {
  __shared__ float sm[256];
  int t = threadIdx.x;
  float s = budget[0] / (sc[S_T] + TINY_);
  unsigned done = ((const unsigned*)sc)[S_DONE];
  int base = blockIdx.x * 1024;
  float dmax = 0.0f;
#pragma unroll
  for (int i = 0; i < 4; ++i) {
    int e = base + i * 256 + t;
    int n = e & (N_ - 1);
    int m = e >> 12;
    float pin = Pin[e];
    float pf = pin * rowscale[m] * colscale[n] * s;
    dmax = fmaxf(dmax, fabsf(pf - pin));
    Pout[e] = done ? pin : pf;
  }
  sm[t] = dmax;
  __syncthreads();
  for (int st = 128; st > 0; st >>= 1) {
    if (t < st) sm[t] = fmaxf(sm[t], sm[t + st]);
    __syncthreads();
  }
  if (t == 0) atomicMax((unsigned*)&sc[S_DELTA], __float_as_uint(sm[0]));
}

// final clipped plan -> d_out; per-block partials of plan*C for the score
__global__ void k_finalwrite(const float* __restrict__ Pin, const float* __restrict__ Cst,
                             float* __restrict__ Pout, const float* __restrict__ rowscale,
                             const float* __restrict__ colscale, float* __restrict__ part) {
  __shared__ float sm[256];
  int t = threadIdx.x;
  int base = blockIdx.x * 1024;
  float acc = 0.0f;
#pragma unroll
  for (int i = 0; i < 4; ++i) {
    int e = base + i * 256 + t;
    int n = e & (N_ - 1);
    int m = e >> 12;
    float pf = Pin[e] * rowscale[m] * colscale[n];
    Pout[e] = pf;
    acc += pf * Cst[e];
  }
  sm[t] = acc;
  __syncthreads();
  for (int st = 128; st > 0; st >>= 1) {
    if (t < st) sm[t] += sm[t + st];
    __syncthreads();
  }
  if (t == 0) part[blockIdx.x] = sm[0];
}

extern "C" void kernel_launch(void* const* d_in, const int* in_sizes, int n_in,
                              void* d_out, int out_size, void* d_ws, size_t ws_size,
                              hipStream_t stream) {
  const float* user = (const float*)d_in[0];    // [1024,128]
  const float* item = (const float*)d_in[1];    // [4096,128]
  const float* a = (const float*)d_in[2];       // source_mass [1024]
  const float* b = (const float*)d_in[3];       // target_capacity [4096]
  const float* budget = (const float*)d_in[4];  // mass_budget [1]
  float* out = (float*)d_out;                   // [score, plan(MN), mass, usage(N)]

  float* ws = (float*)d_ws;
  float* SC = ws + OFF_SC;
  float* rowscale = ws + OFF_ROW;
  float* colscale = ws + OFF_COL;
  float* u2 = ws + OFF_U2;
  float* v2 = ws + OFF_V2;
  float* part = ws + OFF_PART;
  float* Cst = ws + OFF_C;
  float* P0 = ws + OFF_P0;

  float* score = out;
  float* P1 = out + 1;            // plan region doubles as ping-pong buffer
  float* mass = out + 1 + MN;
  float* usage = out + 2 + MN;

  k_init<<<1, 1, 0, stream>>>(SC);
  k_sqnorm<<<(M_ + N_) / 8, 256, 0, stream>>>(user, item, u2, v2);
  k_bmax<<<1, 1024, 0, stream>>>(b, SC);
  k_gemm_cost<<<2048, 256, 0, stream>>>(user, item, u2, v2, Cst, SC);
  k_kernelize<<<4096, 256, 0, stream>>>(Cst, P0, b, SC, part);
  k_reduce4096<<<1, 1024, 0, stream>>>(part, SC + S_SUMK, 1.0f);
  k_scale0<<<4096, 256, 0, stream>>>(P0, SC, budget);

  float* bufs[2] = {P0, P1};
  for (int i = 0; i < ITERS_; ++i) {
    float* in = bufs[i & 1];
    float* o = bufs[(i + 1) & 1];
    k_rowscale<<<1024, 256, 0, stream>>>(in, a, rowscale);
    k_colpart<<<128, 256, 0, stream>>>(in, rowscale, part);
    k_colfinish<<<16, 256, 0, stream>>>(part, b, colscale, nullptr);
    k_tsum<<<1, 1024, 0, stream>>>(part, SC);  // also applies freeze from prev delta
    k_update<<<4096, 256, 0, stream>>>(in, o, rowscale, colscale, SC, budget);
  }

  // ITERS_ even -> converged plan is in P0; final feasibility clip writes into d_out
  float* fin = bufs[ITERS_ & 1];
  k_rowscale<<<1024, 256, 0, stream>>>(fin, a, rowscale);
  k_colpart<<<128, 256, 0, stream>>>(fin, rowscale, part);
  k_colfinish<<<16, 256, 0, stream>>>(part, b, colscale, usage);
  k_reduce4096<<<1, 1024, 0, stream>>>(part, mass, 1.0f);
  k_finalwrite<<<4096, 256, 0, stream>>>(fin, Cst, P1, rowscale, colscale, part);
  k_reduce4096<<<1, 1024, 0, stream>>>(part, score, -1.0f);
}